// GMNet_16114717294805
// MI455X (gfx1250) — compile-verified
//
#include <hip/hip_runtime.h>
#include <math.h>

// ---------------------------------------------------------------------------
// GMNet KKT solve on MI455X (gfx1250, wave32).
//   - M collapses to M[tN+s,uN+v] = [s!=v][t!=u]*(S[s,t]+S[v,u])/2, S = h1 h2^T
//   - K (1368x1368, padded to 1376) solved by unpivoted blocked LU (symmetric
//     quasidefinite => stable without pivoting).
//   - Trailing GEMM updates use V_WMMA_F32_16X16X4_F32: each wave owns a
//     32x32 output tile (2x2 fragments, 16 WMMAs), A/B fragments reused 2x.
//     Tile edge cases are template-specialized so the hot path is branch-free
//     straight-line code (uniform branches would otherwise break load
//     clustering and stall on loadcnt before every WMMA).
//   - fp32 keeps LU accuracy; working set (~61 MB) is L2-resident (192 MB L2).
// ---------------------------------------------------------------------------

#define BB   8
#define NN_  36
#define DD   64
#define NN2  (NN_*NN_)      // 1296
#define NCON (2*NN_)        // 72
#define KSZ  (NN2+NCON)     // 1368
#define PP   1376           // padded to multiple of 16 (and 32)
#define NBLK (PP/16)        // 86
#define KDIAG 1225.0f       // (N-1)^2
#define EPSK  1e-5f
#define ALPHA 1000.0f

typedef float v2f __attribute__((ext_vector_type(2)));
typedef float v8f __attribute__((ext_vector_type(8)));

// ---------------- feature: normalize over node axis, relu(e W^T + b) -------
__global__ void feat_kernel(const float* __restrict__ emb1,
                            const float* __restrict__ emb2,
                            const float* __restrict__ W,
                            const float* __restrict__ bias,
                            float* __restrict__ h1, float* __restrict__ h2) {
    int b = blockIdx.x, which = blockIdx.y;
    const float* e = (which == 0 ? emb1 : emb2) + (size_t)b * NN_ * DD;
    float* h = (which == 0 ? h1 : h2) + (size_t)b * NN_ * DD;
    __shared__ float se[NN_ * DD];
    __shared__ float sw[DD * DD];
    __shared__ float snrm[DD];
    int tid = threadIdx.x;
    for (int i = tid; i < NN_ * DD; i += 256) se[i] = e[i];
    for (int i = tid; i < DD * DD; i += 256) sw[i] = W[i];
    __syncthreads();
    for (int d = tid; d < DD; d += 256) {
        float s = 0.f;
        for (int n = 0; n < NN_; n++) { float v = se[n * DD + d]; s += v * v; }
        snrm[d] = fmaxf(sqrtf(s), 1e-12f);
    }
    __syncthreads();
    for (int i = tid; i < NN_ * DD; i += 256) se[i] = se[i] / snrm[i % DD];
    __syncthreads();
    for (int idx = tid; idx < NN_ * DD; idx += 256) {
        int n = idx / DD, i = idx % DD;
        float acc = bias[i];
        for (int d = 0; d < DD; d++) acc += se[n * DD + d] * sw[i * DD + d];
        h[idx] = fmaxf(acc, 0.f);
    }
}

// ---------------- S = h1 h2^T (36x36 per batch) + rhs assembly -------------
__global__ void gram_rhs_kernel(const float* __restrict__ h1,
                                const float* __restrict__ h2,
                                float* __restrict__ S, float* __restrict__ rhs) {
    int b = blockIdx.x, tid = threadIdx.x;
    __shared__ float s1[NN_ * DD];
    __shared__ float s2[NN_ * DD];
    for (int i = tid; i < NN_ * DD; i += 256) {
        s1[i] = h1[(size_t)b * NN_ * DD + i];
        s2[i] = h2[(size_t)b * NN_ * DD + i];
    }
    __syncthreads();
    for (int idx = tid; idx < NN2; idx += 256) {
        int i = idx / NN_, j = idx % NN_;
        float a = 0.f;
        for (int d = 0; d < DD; d++) a += s1[i * DD + d] * s2[j * DD + d];
        S[(size_t)b * NN2 + idx] = a;
        rhs[(size_t)b * PP + j * NN_ + i] = a;   // rhs[t*N+s] = S[s,t]
    }
    for (int r = NN2 + tid; r < PP; r += 256)
        rhs[(size_t)b * PP + r] = (r < KSZ) ? 1.0f : 0.0f;
}

// ---------------- assemble padded KKT matrix K (1376x1376 per batch) -------
__global__ void assemble_kernel(const float* __restrict__ S, float* __restrict__ K) {
    int b = blockIdx.z;
    int r = blockIdx.y * 16 + threadIdx.y;
    int c = blockIdx.x * 16 + threadIdx.x;
    const float* Sb = S + (size_t)b * NN2;
    float val = 0.f;
    if (r < NN2 && c < NN2) {
        int t = r / NN_, s = r - t * NN_, u = c / NN_, v = c - u * NN_;
        val = (r == c) ? KDIAG : 0.f;
        if (s != v && t != u) val -= 0.5f * (Sb[s * NN_ + t] + Sb[v * NN_ + u]);
    } else if (r < NN2) {              // A^T block
        int i = c - NN2, t = r / NN_, s = r - t * NN_;
        if (i < NN_)            val = (s == i) ? 1.f : 0.f;
        else if (i < NCON)      val = (t == i - NN_) ? 1.f : 0.f;
    } else if (c < NN2) {              // A block
        int i = r - NN2, u = c / NN_, v = c - u * NN_;
        if (i < NN_)            val = (v == i) ? 1.f : 0.f;
        else if (i < NCON)      val = (u == i - NN_) ? 1.f : 0.f;
    } else if (r == c) {
        val = (r < KSZ) ? -EPSK : 1.0f;   // -eps*I, identity on padding
    }
    K[((size_t)b * PP + r) * PP + c] = val;
}

// ---------------- LU: panel factor (16-wide, in dynamic LDS) ---------------
__global__ void lu_panel_kernel(float* __restrict__ K, int kb) {
    extern __shared__ float pan[];           // H x 16
    int b = blockIdx.x, tid = threadIdx.x;
    int r0 = kb * 16, H = PP - r0;
    float* Kb = K + (size_t)b * PP * PP;
    for (int i = tid; i < H * 16; i += 256)
        pan[i] = Kb[(size_t)(r0 + i / 16) * PP + r0 + (i % 16)];
    __syncthreads();
    for (int j = 0; j < 16; j++) {
        float dinv = 1.0f / pan[j * 16 + j];
        for (int i = j + 1 + tid; i < H; i += 256) {
            float lij = pan[i * 16 + j] * dinv;
            pan[i * 16 + j] = lij;
            for (int c2 = j + 1; c2 < 16; c2++)
                pan[i * 16 + c2] -= lij * pan[j * 16 + c2];
        }
        __syncthreads();
    }
    for (int i = tid; i < H * 16; i += 256)
        Kb[(size_t)(r0 + i / 16) * PP + r0 + (i % 16)] = pan[i];
}

// ---------------- LU: U12 = L11^{-1} A12 (unit lower forward solve) --------
__global__ void lu_rowpanel_kernel(float* __restrict__ K, int kb) {
    int b = blockIdx.y, tid = threadIdx.x;
    int r0 = kb * 16;
    int col = r0 + 16 + blockIdx.x * 256 + tid;
    __shared__ float L11[16 * 16];
    float* Kb = K + (size_t)b * PP * PP;
    if (tid < 256)
        L11[tid] = Kb[(size_t)(r0 + tid / 16) * PP + r0 + (tid % 16)];
    __syncthreads();
    if (col < PP) {
        float x[16];
        for (int j = 0; j < 16; j++) x[j] = Kb[(size_t)(r0 + j) * PP + col];
        for (int j = 1; j < 16; j++) {
            float s = x[j];
            for (int t2 = 0; t2 < j; t2++) s -= L11[j * 16 + t2] * x[t2];
            x[j] = s;
        }
        for (int j = 0; j < 16; j++) Kb[(size_t)(r0 + j) * PP + col] = x[j];
    }
}

// ---------------- LU: trailing update A22 -= L21*U12 via fp32 WMMA ---------
// fp32 16x16x4 layouts (wave32): lane L, half=L>>4, l16=L&15.
//   A 16x4 : a[i] = A[m=l16, k=2*half+i]   (two K elements contiguous -> b64)
//   B 4x16 : b[i] = B[k=2*half+i, n=l16]
//   C 16x16: c[r] = C[m=r+8*half, n=l16]
// fp32 WMMA has no A/B NEG modifier (only CNeg) -> negate A on load (v_xor).
// Template-specialized on which fragments exist so the hot 32x32 path is
// branch-free (EXEC all-1s, required for WMMA; guards are wave-uniform).
template <bool R1, bool C1>
__device__ __forceinline__ void update_tile(float* __restrict__ Kb,
                                            int row0, int col0, int kcol,
                                            int lane) {
    int half = lane >> 4, l16 = lane & 15;
    int row1 = row0 + 16, col1 = col0 + 16;

    v8f c00, c01, c10, c11;
    for (int r = 0; r < 8; r++) {
        size_t ro0 = (size_t)(row0 + r + 8 * half) * PP;
        c00[r] = Kb[ro0 + col0 + l16];
        if (C1) c01[r] = Kb[ro0 + col1 + l16];
        if (R1) {
            size_t ro1 = (size_t)(row1 + r + 8 * half) * PP;
            c10[r] = Kb[ro1 + col0 + l16];
            if (C1) c11[r] = Kb[ro1 + col1 + l16];
        }
    }
    for (int kk = 0; kk < 4; kk++) {
        int k0 = kcol + kk * 4;
        int kc = k0 + 2 * half;                     // even element -> 8B aligned
        v2f a0, a1, b0, b1;
        v2f av0 = *(const v2f*)(Kb + (size_t)(row0 + l16) * PP + kc);
        a0[0] = -av0[0]; a0[1] = -av0[1];           // -L21
        if (R1) {
            v2f av1 = *(const v2f*)(Kb + (size_t)(row1 + l16) * PP + kc);
            a1[0] = -av1[0]; a1[1] = -av1[1];
        }
        for (int i = 0; i < 2; i++) {
            size_t kr = (size_t)(kc + i) * PP;
            b0[i] = Kb[kr + col0 + l16];            //  U12
            if (C1) b1[i] = Kb[kr + col1 + l16];
        }
        c00 = __builtin_amdgcn_wmma_f32_16x16x4_f32(false, a0, false, b0, (short)0, c00, false, false);
        if (C1) c01 = __builtin_amdgcn_wmma_f32_16x16x4_f32(false, a0, false, b1, (short)0, c01, false, false);
        if (R1) c10 = __builtin_amdgcn_wmma_f32_16x16x4_f32(false, a1, false, b0, (short)0, c10, false, false);
        if (R1 && C1) c11 = __builtin_amdgcn_wmma_f32_16x16x4_f32(false, a1, false, b1, (short)0, c11, false, false);
    }
    for (int r = 0; r < 8; r++) {
        size_t ro0 = (size_t)(row0 + r + 8 * half) * PP;
        Kb[ro0 + col0 + l16] = c00[r];
        if (C1) Kb[ro0 + col1 + l16] = c01[r];
        if (R1) {
            size_t ro1 = (size_t)(row1 + r + 8 * half) * PP;
            Kb[ro1 + col0 + l16] = c10[r];
            if (C1) Kb[ro1 + col1 + l16] = c11[r];
        }
    }
}

__global__ void lu_update_kernel(float* __restrict__ K, int kb) {
    int b = blockIdx.z;
    float* Kb = K + (size_t)b * PP * PP;
    int base = kb * 16 + 16;
    int row0 = base + blockIdx.y * 32;
    int col0 = base + blockIdx.x * 32;
    bool hasR1 = (row0 + 16 < PP), hasC1 = (col0 + 16 < PP);
    int lane = threadIdx.x;
    int kcol = kb * 16;
    if (hasR1 && hasC1)      update_tile<true,  true >(Kb, row0, col0, kcol, lane);
    else if (hasR1)          update_tile<true,  false>(Kb, row0, col0, kcol, lane);
    else if (hasC1)          update_tile<false, true >(Kb, row0, col0, kcol, lane);
    else                     update_tile<false, false>(Kb, row0, col0, kcol, lane);
}

// ---------------- triangular solves: L y = rhs ; U x = y -------------------
__global__ void trisolve_kernel(const float* __restrict__ K,
                                const float* __restrict__ rhs,
                                float* __restrict__ x) {
    int b = blockIdx.x, tid = threadIdx.x;
    __shared__ float y[PP];
    const float* Kb = K + (size_t)b * PP * PP;
    for (int i = tid; i < PP; i += 256) y[i] = rhs[(size_t)b * PP + i];
    __syncthreads();
    for (int kb = 0; kb < NBLK; kb++) {          // forward, unit lower
        int r0 = kb * 16;
        if (tid == 0) {
            for (int j = 1; j < 16; j++) {
                float s = y[r0 + j];
                for (int t2 = 0; t2 < j; t2++)
                    s -= Kb[(size_t)(r0 + j) * PP + r0 + t2] * y[r0 + t2];
                y[r0 + j] = s;
            }
        }
        __syncthreads();
        for (int i = r0 + 16 + tid; i < PP; i += 256) {
            float s = y[i];
            for (int j = 0; j < 16; j++)
                s -= Kb[(size_t)i * PP + r0 + j] * y[r0 + j];
            y[i] = s;
        }
        __syncthreads();
    }
    for (int kb = NBLK - 1; kb >= 0; kb--) {     // backward, upper
        int r0 = kb * 16;
        if (tid == 0) {
            for (int j = 15; j >= 0; j--) {
                float s = y[r0 + j];
                for (int t2 = j + 1; t2 < 16; t2++)
                    s -= Kb[(size_t)(r0 + j) * PP + r0 + t2] * y[r0 + t2];
                y[r0 + j] = s / Kb[(size_t)(r0 + j) * PP + r0 + j];
            }
        }
        __syncthreads();
        for (int i = tid; i < r0; i += 256) {
            float s = y[i];
            for (int j = 0; j < 16; j++)
                s -= Kb[(size_t)i * PP + r0 + j] * y[r0 + j];
            y[i] = s;
        }
        __syncthreads();
    }
    for (int i = tid; i < PP; i += 256) x[(size_t)b * PP + i] = y[i];
}

// ---------------- transpose + clamp + softmax(alpha*s) ---------------------
__global__ void vote_kernel(const float* __restrict__ x, float* __restrict__ out) {
    int s = blockIdx.x, b = blockIdx.y, tid = threadIdx.x;
    __shared__ float vals[NN_];
    if (tid < NN_) {
        float v = x[(size_t)b * PP + tid * NN_ + s];   // transpose of reshape
        vals[tid] = fminf(fmaxf(v, 0.f), 1.f);
    }
    __syncthreads();
    if (tid == 0) {
        float m = -1e30f;
        for (int t = 0; t < NN_; t++) m = fmaxf(m, vals[t]);
        float sum = 0.f;
        for (int t = 0; t < NN_; t++) { float e = __expf(ALPHA * (vals[t] - m)); vals[t] = e; sum += e; }
        float inv = 1.f / sum;
        for (int t = 0; t < NN_; t++) out[((size_t)b * NN_ + s) * NN_ + t] = vals[t] * inv;
    }
}

// ---------------------------------------------------------------------------
extern "C" void kernel_launch(void* const* d_in, const int* in_sizes, int n_in,
                              void* d_out, int out_size, void* d_ws, size_t ws_size,
                              hipStream_t stream) {
    const float* emb1 = (const float*)d_in[0];
    const float* emb2 = (const float*)d_in[1];
    const float* Wcg  = (const float*)d_in[2];
    const float* bcg  = (const float*)d_in[3];

    float* ws  = (float*)d_ws;
    size_t kelem = (size_t)BB * PP * PP;
    float* Km  = ws;                       // 8*1376*1376 floats (~60.6 MB)
    float* h1  = Km + kelem;
    float* h2  = h1 + (size_t)BB * NN_ * DD;
    float* S   = h2 + (size_t)BB * NN_ * DD;
    float* rhs = S  + (size_t)BB * NN2;
    float* x   = rhs + (size_t)BB * PP;

    feat_kernel<<<dim3(BB, 2), 256, 0, stream>>>(emb1, emb2, Wcg, bcg, h1, h2);
    gram_rhs_kernel<<<BB, 256, 0, stream>>>(h1, h2, S, rhs);
    assemble_kernel<<<dim3(NBLK, NBLK, BB), dim3(16, 16), 0, stream>>>(S, Km);

    for (int kb = 0; kb < NBLK; kb++) {
        int H = PP - kb * 16;
        lu_panel_kernel<<<BB, 256, (size_t)H * 16 * sizeof(float), stream>>>(Km, kb);
        int W2 = PP - (kb + 1) * 16;
        if (W2 > 0) {
            lu_rowpanel_kernel<<<dim3((W2 + 255) / 256, BB), 256, 0, stream>>>(Km, kb);
            int tiles = W2 / 16;
            int t32 = (tiles + 1) / 2;
            lu_update_kernel<<<dim3(t32, t32, BB), 32, 0, stream>>>(Km, kb);
        }
    }

    trisolve_kernel<<<BB, 256, 0, stream>>>(Km, rhs, x);
    vote_kernel<<<dim3(NN_, BB), 64, 0, stream>>>(x, (float*)d_out);
}